// GlobalBlock_24807731101813
// MI455X (gfx1250) — compile-verified
//
#include <hip/hip_runtime.h>
#include <hip/hip_bf16.h>

typedef __attribute__((ext_vector_type(2))) float v2f;
typedef __attribute__((ext_vector_type(4))) float v4f;
typedef __attribute__((ext_vector_type(8))) float v8f;

#define NB_E 512   // partial-sum blocks for edge_attributes
#define NB_N 64    // partial-sum blocks for node_attributes

// ---------------------------------------------------------------------------
// Pass 1: column sums. Block b < NB_E sums a row-range of edges; blocks
// [NB_E, NB_E+NB_N) sum a row-range of nodes. Each block writes its own
// 64-float partial to ws[b*64..] (block-owned slots -> deterministic, no
// atomics, no ws init needed).
// Thread layout: cg = tid&15 selects the float4 column group, rl = tid>>4 is
// a row slot; a wave loads 512 contiguous bytes per step. Two independent
// accumulator chains keep >=4 b128 loads in flight; loads are non-temporal
// (stream is read exactly once, don't pollute the 192MB L2).
// ---------------------------------------------------------------------------
__global__ __launch_bounds__(256)
void reduce_cols(const float* __restrict__ edges, const float* __restrict__ nodes,
                 float* __restrict__ partial, int E, int N) {
    int b = blockIdx.x;
    const float* src;
    int rows, nb, bl;
    if (b < NB_E) { src = edges; rows = E; nb = NB_E; bl = b; }
    else          { src = nodes; rows = N; nb = NB_N; bl = b - NB_E; }

    int per = (rows + nb - 1) / nb;
    int r0  = bl * per;
    int r1  = r0 + per; if (r1 > rows) r1 = rows;

    int tid = threadIdx.x;
    int cg  = tid & 15;    // float4 column group: cols [cg*4, cg*4+4)
    int rl  = tid >> 4;    // row slot 0..15

    v4f acc0 = {0.f, 0.f, 0.f, 0.f};
    v4f acc1 = {0.f, 0.f, 0.f, 0.f};

    int r = r0 + rl;
    #pragma unroll 2
    for (; r + 16 < r1; r += 32) {
        acc0 += __builtin_nontemporal_load(
            reinterpret_cast<const v4f*>(src + (size_t)r * 64 + cg * 4));
        acc1 += __builtin_nontemporal_load(
            reinterpret_cast<const v4f*>(src + (size_t)(r + 16) * 64 + cg * 4));
    }
    if (r < r1) {
        acc0 += __builtin_nontemporal_load(
            reinterpret_cast<const v4f*>(src + (size_t)r * 64 + cg * 4));
    }
    acc0 += acc1;

    __shared__ v4f s[256];
    s[tid] = acc0;
    __syncthreads();
    // tree-reduce over the 16 row slots (tid = rl*16 + cg)
    for (int stride = 128; stride >= 16; stride >>= 1) {
        if (tid < stride) s[tid] += s[tid + stride];
        __syncthreads();
    }
    if (tid < 16) {
        reinterpret_cast<v4f*>(partial + (size_t)blockIdx.x * 64)[tid] = s[tid];
    }
}

// ---------------------------------------------------------------------------
// Pass 2 (single block, 16 waves): reduce partials -> feats[192], then
// 2-layer MLP via V_WMMA_F32_16X16X4_F32. The 1x192 activation is lifted to a
// 16xK A-matrix with rows 1..15 zero (zeroed by multiplying with a 0/1 lane
// mask -> plain ds_load + v_mul, no EXEC manipulation), so D row M=0 is the
// result row.
//   A layout (16x4 f32): lanes 0-15 -> M=lane, a.x=K0 a.y=K1;
//                        lanes 16-31 -> M=lane-16, a.x=K2 a.y=K3.
//   B layout (4x16 f32): VGPR0 lanes 0-15 = row K0, lanes 16-31 = row K1;
//                        VGPR1 lanes 0-15 = row K2, lanes 16-31 = row K3.
//   C/D (16x16 f32): c[0] = row M=0 on lanes 0-15 (N=lane).
// All branches around WMMA are wave-uniform => EXEC all-ones (ISA req).
// ---------------------------------------------------------------------------
__global__ __launch_bounds__(512)
void finalize(const float* __restrict__ partial, const float* __restrict__ g,
              const float* __restrict__ W1, const float* __restrict__ b1,
              const float* __restrict__ W2, const float* __restrict__ b2,
              float* __restrict__ out, int E, int N) {
    __shared__ float feats[192];
    __shared__ float hbuf[256];
    __shared__ float red[512];

    int tid = threadIdx.x;
    int col = tid & 63;      // 0..63
    int chk = tid >> 6;      // 0..7

    // ---- reduce the 512 edge partials + 64 node partials ----
    float se = 0.f;
    for (int i = chk; i < NB_E; i += 8) se += partial[(size_t)i * 64 + col];
    float sn = 0.f;
    for (int i = chk; i < NB_N; i += 8) sn += partial[(size_t)(NB_E + i) * 64 + col];

    red[tid] = se;
    __syncthreads();
    if (chk == 0) {
        float t = 0.f;
        #pragma unroll
        for (int c2 = 0; c2 < 8; ++c2) t += red[c2 * 64 + col];
        feats[col] = t / (float)E;                    // mean over edges
    }
    __syncthreads();
    red[tid] = sn;
    __syncthreads();
    if (chk == 0) {
        float t = 0.f;
        #pragma unroll
        for (int c2 = 0; c2 < 8; ++c2) t += red[c2 * 64 + col];
        feats[64 + col] = t / (float)N;               // mean over nodes
        feats[128 + col] = g[col];                    // global attrs
    }
    __syncthreads();

    int wave = tid >> 5;       // 0..15
    int lane = tid & 31;
    int m    = lane & 15;      // A-matrix M index (only M==0 carries data)
    int hiK  = (lane >= 16) ? 1 : 0;
    float msel = (m == 0) ? 1.f : 0.f;   // lane mask: zero rows 1..15 of A

    // ---- layer 1: h = relu(feats @ W1 + b1), wave w owns cols [16w,16w+16) ----
    {
        v8f c = {};
        for (int k0 = 0; k0 < 192; k0 += 4) {
            float f0 = feats[k0 + 2 * hiK];          // unconditional ds_load
            float f1 = feats[k0 + 2 * hiK + 1];
            v2f a;
            a.x = f0 * msel;
            a.y = f1 * msel;
            v2f bb;
            bb.x = W1[(size_t)(k0 + hiK)     * 256 + wave * 16 + m];
            bb.y = W1[(size_t)(k0 + 2 + hiK) * 256 + wave * 16 + m];
            c = __builtin_amdgcn_wmma_f32_16x16x4_f32(
                    false, a, false, bb, (short)0, c, false, false);
        }
        if (lane < 16) {
            int j = wave * 16 + lane;
            float hv = c[0] + b1[j];
            hbuf[j] = hv > 0.f ? hv : 0.f;
        }
    }
    __syncthreads();

    // ---- layer 2: out = h @ W2 + b2, waves 0..3 own cols [16w,16w+16) ----
    if (wave < 4) {
        v8f c = {};
        for (int k0 = 0; k0 < 256; k0 += 4) {
            float f0 = hbuf[k0 + 2 * hiK];
            float f1 = hbuf[k0 + 2 * hiK + 1];
            v2f a;
            a.x = f0 * msel;
            a.y = f1 * msel;
            v2f bb;
            bb.x = W2[(size_t)(k0 + hiK)     * 64 + wave * 16 + m];
            bb.y = W2[(size_t)(k0 + 2 + hiK) * 64 + wave * 16 + m];
            c = __builtin_amdgcn_wmma_f32_16x16x4_f32(
                    false, a, false, bb, (short)0, c, false, false);
        }
        if (lane < 16) {
            int j = wave * 16 + lane;
            out[j] = c[0] + b2[j];
        }
    }
}

extern "C" void kernel_launch(void* const* d_in, const int* in_sizes, int n_in,
                              void* d_out, int out_size, void* d_ws, size_t ws_size,
                              hipStream_t stream) {
    const float* edges = (const float*)d_in[0];   // [E,64]
    const float* nodes = (const float*)d_in[1];   // [N,64]
    const float* g     = (const float*)d_in[2];   // [64]
    const float* W1    = (const float*)d_in[3];   // [192,256]
    const float* b1    = (const float*)d_in[4];   // [256]
    const float* W2    = (const float*)d_in[5];   // [256,64]
    const float* b2    = (const float*)d_in[6];   // [64]
    float* out = (float*)d_out;                   // [64]

    int E = in_sizes[0] / 64;
    int N = in_sizes[1] / 64;
    float* partial = (float*)d_ws;                // (NB_E+NB_N)*64 floats = 147 KB

    reduce_cols<<<NB_E + NB_N, 256, 0, stream>>>(edges, nodes, partial, E, N);
    finalize<<<1, 512, 0, stream>>>(partial, g, W1, b1, W2, b2, out, E, N);
}